// GraphAttention_36541581754849
// MI455X (gfx1250) — compile-verified
//
#include <hip/hip_runtime.h>
#include <hip/hip_bf16.h>

#define IN_FEAT 128
#define UNITS   128
#define EPS     1e-7f
#define SLOPE   0.2f

typedef __attribute__((ext_vector_type(2))) float v2f;
typedef __attribute__((ext_vector_type(8))) float v8f;

// -------------------------------------------------------------------------
// Zero-fill (out and denom must be re-zeroed every call: atomics accumulate)
// -------------------------------------------------------------------------
__global__ void gat_zero_f32(float* __restrict__ p, int n) {
    int i = blockIdx.x * blockDim.x + threadIdx.x;
    if (i < n) p[i] = 0.0f;
}

// -------------------------------------------------------------------------
// H = X @ W via V_WMMA_F32_16X16X4_F32.
// One wave computes a full 16x128 row strip of H: A (16x4) is loaded once
// per k-step and reused across 8 column tiles -> 8 consecutive WMMAs.
// Layouts (wave32):
//   A (16x4 f32, 2 VGPRs): lane l -> M = l%16, K = 2*(l/16) + {0,1}
//   B (4x16 f32, 2 VGPRs): lane l -> N = l%16, K = 2*(l/16) + {0,1}
//   C/D (16x16 f32, 8 VGPRs): vgpr i, lane l -> M = i + 8*(l/16), N = l%16
// EXEC is all-ones at every WMMA: the strip guard is wave-uniform and the
// row clamp is a branchless select.
// -------------------------------------------------------------------------
__global__ void gat_gemm_wmma(const float* __restrict__ X,   // n_nodes x 128
                              const float* __restrict__ W,   // 128 x 128
                              float* __restrict__ H,         // n_nodes x 128
                              int n_nodes) {
    const int lane  = threadIdx.x & 31;
    const int strip = blockIdx.x * (blockDim.x >> 5) + (threadIdx.x >> 5);
    const int row0  = strip * 16;
    if (row0 >= n_nodes) return;          // wave-uniform exit

    const int m     = lane & 15;          // M for A, N for B/C
    const int khalf = (lane >> 4) << 1;   // 0 or 2

    int r = row0 + m;
    if (r > n_nodes - 1) r = n_nodes - 1; // clamp (padded rows never stored)
    const float* __restrict__ xrow = X + (size_t)r * IN_FEAT;

    v8f acc[8] = {};
    for (int k = 0; k < IN_FEAT; k += 4) {
        v2f a;
        a.x = xrow[k + khalf + 0];
        a.y = xrow[k + khalf + 1];
        const float* __restrict__ w0 = W + (k + khalf) * UNITS + m;
        #pragma unroll
        for (int t = 0; t < 8; ++t) {
            v2f b;
            b.x = w0[t * 16];
            b.y = w0[UNITS + t * 16];
            // 8 args: (neg_a, A, neg_b, B, c_mod, C, reuse_a, reuse_b)
            acc[t] = __builtin_amdgcn_wmma_f32_16x16x4_f32(
                false, a, false, b, (short)0, acc[t], false, false);
        }
    }

    const int halfOff = (lane >> 4) << 3; // 0 or 8
    if (row0 + 15 < n_nodes) {
        // full strip: unguarded stores
        float* __restrict__ hp = H + (size_t)(row0 + halfOff) * UNITS + m;
        #pragma unroll
        for (int t = 0; t < 8; ++t) {
            #pragma unroll
            for (int i = 0; i < 8; ++i) {
                hp[(size_t)i * UNITS + t * 16] = acc[t][i];
            }
        }
    } else {
        #pragma unroll
        for (int t = 0; t < 8; ++t) {
            #pragma unroll
            for (int i = 0; i < 8; ++i) {
                int rr = row0 + i + halfOff;
                if (rr < n_nodes) H[(size_t)rr * UNITS + t * 16 + m] = acc[t][i];
            }
        }
    }
}

// -------------------------------------------------------------------------
// Per-node attention pre-scores:
//   s0[n] = h[n] . ka[0:128]   (target half)
//   s1[n] = h[n] . ka[128:256] (source half)
// One wave per node, 4 elems per lane, wave shuffle reduction.
// -------------------------------------------------------------------------
__global__ void gat_node_scores(const float* __restrict__ H,
                                const float* __restrict__ KA,  // 256 x 1
                                float* __restrict__ s0,
                                float* __restrict__ s1,
                                int n_nodes) {
    const int node = blockIdx.x * (blockDim.x >> 5) + (threadIdx.x >> 5);
    const int lane = threadIdx.x & 31;
    if (node >= n_nodes) return;

    const float* __restrict__ hr = H + (size_t)node * UNITS;
    float a0 = 0.0f, a1 = 0.0f;
    #pragma unroll
    for (int j = lane; j < UNITS; j += 32) {
        float hv = hr[j];
        a0 = fmaf(hv, KA[j], a0);
        a1 = fmaf(hv, KA[UNITS + j], a1);
    }
    #pragma unroll
    for (int off = 16; off > 0; off >>= 1) {
        a0 += __shfl_down(a0, off, 32);
        a1 += __shfl_down(a1, off, 32);
    }
    if (lane == 0) { s0[node] = a0; s1[node] = a1; }
}

// -------------------------------------------------------------------------
// Edge pass 1: score -> leaky_relu -> clip -> exp; accumulate denom[tgt].
// -------------------------------------------------------------------------
__global__ void gat_edge_scores(const int* __restrict__ edges,  // n_edges x 2
                                const float* __restrict__ s0,
                                const float* __restrict__ s1,
                                float* __restrict__ wts,
                                float* __restrict__ denom,
                                int n_edges) {
    const int e = blockIdx.x * blockDim.x + threadIdx.x;
    if (e >= n_edges) return;
    const int tgt = edges[2 * e + 0];
    const int src = edges[2 * e + 1];
    float sc = s0[tgt] + s1[src];
    sc = (sc >= 0.0f) ? sc : SLOPE * sc;                 // leaky_relu
    sc = fminf(2.0f, fmaxf(-2.0f, sc));                  // clip
    float w = expf(sc);
    wts[e] = w;
    atomicAdd(&denom[tgt], w);
}

// -------------------------------------------------------------------------
// Edge pass 2: out[tgt] += (w / (denom[tgt]+eps)) * h[src]
// One wave per edge: 128-float row = float4 per lane + 4 f32 atomics.
// -------------------------------------------------------------------------
__global__ void gat_edge_aggregate(const int* __restrict__ edges,
                                   const float* __restrict__ H,
                                   const float* __restrict__ wts,
                                   const float* __restrict__ denom,
                                   float* __restrict__ out,
                                   int n_edges) {
    const int e    = blockIdx.x * (blockDim.x >> 5) + (threadIdx.x >> 5);
    const int lane = threadIdx.x & 31;
    if (e >= n_edges) return;

    const int tgt = edges[2 * e + 0];
    const int src = edges[2 * e + 1];

    const float* __restrict__ hrow = H + (size_t)src * UNITS;
    __builtin_prefetch(hrow + lane * 4, 0, 0);           // global_prefetch_b8

    const float alpha = wts[e] / (denom[tgt] + EPS);
    const float4 hv = ((const float4*)hrow)[lane];
    float* __restrict__ o = out + (size_t)tgt * UNITS + lane * 4;
    atomicAdd(o + 0, alpha * hv.x);
    atomicAdd(o + 1, alpha * hv.y);
    atomicAdd(o + 2, alpha * hv.z);
    atomicAdd(o + 3, alpha * hv.w);
}

// -------------------------------------------------------------------------
extern "C" void kernel_launch(void* const* d_in, const int* in_sizes, int n_in,
                              void* d_out, int out_size, void* d_ws, size_t ws_size,
                              hipStream_t stream) {
    const float* node_states = (const float*)d_in[0];
    const int*   edges       = (const int*)d_in[1];
    const float* kernelW     = (const float*)d_in[2];
    const float* kernelAtt   = (const float*)d_in[3];
    float*       out         = (float*)d_out;

    const int n_nodes = in_sizes[0] / IN_FEAT;
    const int n_edges = in_sizes[1] / 2;

    // Workspace layout
    float* H     = (float*)d_ws;                       // n_nodes * 128
    float* s0    = H + (size_t)n_nodes * UNITS;        // n_nodes
    float* s1    = s0 + n_nodes;                       // n_nodes
    float* wts   = s1 + n_nodes;                       // n_edges
    float* denom = wts + n_edges;                      // n_nodes

    // Zero accumulators (every call: atomics accumulate across graph replays)
    {
        int n = n_nodes * UNITS;
        gat_zero_f32<<<(n + 255) / 256, 256, 0, stream>>>(out, n);
        gat_zero_f32<<<(n_nodes + 255) / 256, 256, 0, stream>>>(denom, n_nodes);
    }

    // H = X @ W  (WMMA f32 16x16x4; one wave per 16x128 row strip)
    {
        int strips = (n_nodes + 15) / 16;
        int blocks = (strips + 7) / 8;                 // 8 waves per block
        gat_gemm_wmma<<<blocks, 256, 0, stream>>>(node_states, kernelW, H, n_nodes);
    }

    // Per-node pre-scores
    gat_node_scores<<<(n_nodes + 7) / 8, 256, 0, stream>>>(H, kernelAtt, s0, s1, n_nodes);

    // Edge softmax numerators + denominators
    gat_edge_scores<<<(n_edges + 255) / 256, 256, 0, stream>>>(edges, s0, s1, wts, denom, n_edges);

    // Weighted aggregation
    gat_edge_aggregate<<<(n_edges + 7) / 8, 256, 0, stream>>>(edges, H, wts, denom, out, n_edges);
}